// LIF_8400956031530
// MI455X (gfx1250) — compile-verified
//
#include <hip/hip_runtime.h>
#include <cstdint>
#include <cstddef>

// LIF spiking-neuron scan for MI455X (gfx1250).
// Memory-bound streaming recurrence over T=64: depth-8 async global->LDS
// pipeline (ASYNCcnt-tracked, zero VGPR cost for in-flight data) hides HBM
// latency inside the serial time loop; b128 accesses; non-temporal stores
// for the 256MB write-once output.

typedef float v4f __attribute__((ext_vector_type(4)));

#define TPB   256
#define VEC   4
#define DEPTH 8

constexpr int kB = 32, kT = 64, kN = 16384;

__device__ __forceinline__ void async_tile_load(uint32_t lds_addr, const float* gaddr) {
  // GLOBAL_LOAD_ASYNC_TO_LDS_B128: per-lane 16B global->LDS copy, ASYNCcnt-tracked.
  asm volatile("global_load_async_to_lds_b128 %0, %1, off"
               :: "v"(lds_addr), "v"(gaddr)
               : "memory");
}

__device__ __forceinline__ void wait_async_le(int pending) {
  // `pending` is a compile-time constant at every call site (folded by unroll).
  switch (pending) {
    case 0:  asm volatile("s_wait_asynccnt 0" ::: "memory"); break;
    case 1:  asm volatile("s_wait_asynccnt 1" ::: "memory"); break;
    case 2:  asm volatile("s_wait_asynccnt 2" ::: "memory"); break;
    case 3:  asm volatile("s_wait_asynccnt 3" ::: "memory"); break;
    case 4:  asm volatile("s_wait_asynccnt 4" ::: "memory"); break;
    case 5:  asm volatile("s_wait_asynccnt 5" ::: "memory"); break;
    case 6:  asm volatile("s_wait_asynccnt 6" ::: "memory"); break;
    default: asm volatile("s_wait_asynccnt 7" ::: "memory"); break;
  }
}

__global__ __launch_bounds__(TPB) void lif_scan_kernel(
    const float* __restrict__ x,
    const float* __restrict__ thresh,
    const float* __restrict__ tau_x,
    float* __restrict__ spikes,
    float* __restrict__ mems)
{
  __shared__ __align__(16) float tile[DEPTH][TPB * VEC];   // 8 x 4KB ring

  const int tid = threadIdx.x;
  constexpr int nPerBlk  = TPB * VEC;       // 1024 neurons per block
  constexpr int blksPerB = kN / nPerBlk;    // 16
  const int b  = blockIdx.x / blksPerB;
  const int n0 = (blockIdx.x % blksPerB) * nPerBlk + tid * VEC;

  // Per-neuron constants (loaded once; sigmoid via v_exp_f32).
  const v4f thv = *(const v4f*)(thresh + n0);
  const v4f txv = *(const v4f*)(tau_x  + n0);
  v4f sig;
  #pragma unroll
  for (int k = 0; k < VEC; ++k) sig[k] = 1.0f / (1.0f + __expf(-txv[k]));

  const size_t base = (size_t)b * kT * kN + (size_t)n0;
  const float* gx = x      + base;
  float*       gs = spikes + base;
  float*       gm = mems   + base;

  // Low 32 bits of a flat shared pointer == LDS byte offset (aperture rule).
  const uint32_t lds0 = (uint32_t)(uintptr_t)(&tile[0][tid * VEC]);
  constexpr uint32_t ldsStride = nPerBlk * (uint32_t)sizeof(float);   // 4096

  // Prologue: fill the 8-deep async pipeline (t = 0..7).
  #pragma unroll
  for (int p = 0; p < DEPTH; ++p)
    async_tile_load(lds0 + (uint32_t)p * ldsStride, gx + (size_t)p * kN);

  v4f mem = {0.f, 0.f, 0.f, 0.f};

  // One LIF timestep: integrate, fire, hard-reset, emit spike+mem.
  auto step = [&](int t, v4f xv) {
    v4f spk, mo;
    #pragma unroll
    for (int k = 0; k < VEC; ++k) {
      float m = mem[k];
      m = m + sig[k] * (xv[k] - m);        // leaky integration (fma)
      const bool fire = (m >= thv[k]);     // mem - thresh >= 0
      spk[k] = fire ? 1.0f : 0.0f;
      m      = fire ? 0.0f : m;            // hard reset
      mem[k] = m;
      mo[k]  = m;
    }
    __builtin_nontemporal_store(spk, (v4f*)(gs + (size_t)t * kN));
    __builtin_nontemporal_store(mo,  (v4f*)(gm + (size_t)t * kN));
  };

  // Steady state: branch-free, DEPTH loads always in flight.
  #pragma unroll DEPTH
  for (int t = 0; t < kT - DEPTH; ++t) {
    wait_async_le(DEPTH - 1);                         // tile t landed
    const int slot = t & (DEPTH - 1);
    v4f xv = *(const v4f*)(&tile[slot][tid * VEC]);   // ds_load_b128
    // Retire the LDS read before the async engine may overwrite this slot.
    asm volatile("s_wait_dscnt 0" ::: "memory");
    async_tile_load(lds0 + (uint32_t)slot * ldsStride,
                    gx + (size_t)(t + DEPTH) * kN);
    step(t, xv);
  }

  // Drain: last DEPTH tiles, exact wait thresholds, no new issues.
  #pragma unroll
  for (int j = 0; j < DEPTH; ++j) {
    const int t = kT - DEPTH + j;
    wait_async_le(DEPTH - 1 - j);
    const int slot = t & (DEPTH - 1);
    v4f xv = *(const v4f*)(&tile[slot][tid * VEC]);
    step(t, xv);
  }
}

extern "C" void kernel_launch(void* const* d_in, const int* in_sizes, int n_in,
                              void* d_out, int out_size, void* d_ws, size_t ws_size,
                              hipStream_t stream) {
  const float* x      = (const float*)d_in[0];
  const float* thresh = (const float*)d_in[1];
  const float* tau_x  = (const float*)d_in[2];

  float* spikes = (float*)d_out;
  float* mems   = spikes + (size_t)kB * kT * kN;

  const int grid = (kB * kN) / (TPB * VEC);   // 512 blocks, 8 waves each
  lif_scan_kernel<<<grid, TPB, 0, stream>>>(x, thresh, tau_x, spikes, mems);
}